// PrimalDualNetwork_74363063763593
// MI455X (gfx1250) — compile-verified
//
#include <hip/hip_runtime.h>
#include <stdint.h>

// Chambolle-Pock primal-dual ROF, 4096x4096, 20 iterations.
// Memory-bound stencil: use gfx1250 async global->LDS copies for tile staging.

#define MM 4096
#define NN 4096
constexpr int   TS      = 32;                       // tile size (32x32 interior per block)
constexpr int   MAXIT   = 20;
constexpr float SIGMA_C = 1.0f / (7.0f * 0.01f);
constexpr float TAU_C   = 0.01f;
constexpr float LT_C    = 7.0f * 0.01f;             // lambda * tau
constexpr float RDEN_C  = 1.0f / (1.0f + 7.0f * 0.01f);

// ---------- gfx1250 async global -> LDS copy (per-lane B32 gather) ----------
// VDST = per-lane LDS byte address (hardware adds LDS_BASE), VADDR = 64-bit
// global address. Tracked by ASYNCcnt. Flat-address low 32 bits == LDS offset.
__device__ __forceinline__ void async_g2l_b32(const float* gaddr, float* lds_flat) {
    uint32_t loff = (uint32_t)(uintptr_t)lds_flat;
    asm volatile("global_load_async_to_lds_b32 %0, %1, off"
                 :: "v"(loff), "v"(gaddr)
                 : "memory");
}
__device__ __forceinline__ void wait_async_zero() {
    asm volatile("s_wait_asynccnt 0" ::: "memory");
}

__device__ __forceinline__ int clampi(int v, int lo, int hi) {
    return v < lo ? lo : (v > hi ? hi : v);
}

// =====================  Fully fused iteration (6-plane ws)  =====================
// One launch per outer iteration. Each 32x32 tile:
//   A) async-stage x_tilde (34x34, clamp-replicated) into LDS
//   B) compute dual update y_new over the 33x33 extended region (top/left halo
//      recomputed), keep h = w0*y0n, v = w1*y1n in LDS; write interior y_new
//   C) primal update + over-relaxation from LDS h/v.
__global__ __launch_bounds__(1024)
void pd_fused(const float* __restrict__ img,
              const float* __restrict__ w,       // 2 planes
              const float* __restrict__ xt_src,
              const float* __restrict__ y_src,   // 2 planes
              const float* __restrict__ x_src,
              float* __restrict__ y_dst,         // 2 planes
              float* __restrict__ x_dst,
              float* __restrict__ xt_dst)
{
    constexpr int EX = TS + 1;   // 33: extended (dual halo) region
    constexpr int SX = TS + 2;   // 34: x_tilde staging region
    __shared__ float XT[SX * SX];
    __shared__ float H[EX * EX];
    __shared__ float V[EX * EX];

    const int tx  = threadIdx.x, ty = threadIdx.y;
    const int tid = ty * TS + tx;
    const int gx0 = blockIdx.x * TS;
    const int gy0 = blockIdx.y * TS;
    const size_t MN = (size_t)MM * NN;

    // --- Phase A: async stage x_tilde rows [gy0-1, gy0+TS], cols [gx0-1, gx0+TS]
    for (int e = tid; e < SX * SX; e += TS * TS) {
        int sy = e / SX, sx = e - sy * SX;
        int gy = clampi(gy0 - 1 + sy, 0, MM - 1);
        int gx = clampi(gx0 - 1 + sx, 0, NN - 1);
        async_g2l_b32(xt_src + (size_t)gy * NN + gx, &XT[e]);
    }
    wait_async_zero();
    __syncthreads();

    // --- Phase B: dual update over extended region (ey,ex) <-> (gy0-1+ey, gx0-1+ex)
    for (int e = tid; e < EX * EX; e += TS * TS) {
        int ey = e / EX, ex = e - ey * EX;
        int gy = gy0 - 1 + ey;                 // in [-1, 4095]
        int gx = gx0 - 1 + ex;
        int rg = gy < 0 ? 0 : gy;
        int cg = gx < 0 ? 0 : gx;
        size_t off = (size_t)rg * NN + cg;
        float w0  = w[off];
        float w1  = w[MN + off];
        float y0v = y_src[off];
        float y1v = y_src[MN + off];
        float xc  = XT[ey * SX + ex];
        float xr  = XT[ey * SX + ex + 1];      // clamp-replicate => 0 grad at j=N-1
        float xd  = XT[(ey + 1) * SX + ex];    // clamp-replicate => 0 grad at i=M-1
        float y0n = y0v + SIGMA_C * (xr - xc) * w0;
        float y1n = y1v + SIGMA_C * (xd - xc) * w1;
        y0n = fminf(fmaxf(y0n, -1.0f), 1.0f);  // prox: project onto L-inf ball
        y1n = fminf(fmaxf(y1n, -1.0f), 1.0f);
        H[e] = w0 * y0n;
        V[e] = w1 * y1n;
        if (ey >= 1 && ex >= 1) {              // interior pixels owned by this tile
            size_t o2 = (size_t)gy * NN + gx;
            y_dst[o2]      = y0n;
            y_dst[MN + o2] = y1n;
        }
    }
    __syncthreads();

    // --- Phase C: primal + over-relaxation for interior pixel (i,j)
    const int i = gy0 + ty, j = gx0 + tx;
    const size_t o = (size_t)i * NN + j;
    float hC = H[(ty + 1) * EX + tx + 1];      // h(i, j)
    float hL = H[(ty + 1) * EX + tx];          // h(i, j-1)
    float vC = V[(ty + 1) * EX + tx + 1];      // v(i, j)
    float vU = V[ty * EX + tx + 1];            // v(i-1, j)
    float dh = (j < NN - 1 ? hC : 0.0f) - (j > 0 ? hL : 0.0f);
    float dv = (i < MM - 1 ? vC : 0.0f) - (i > 0 ? vU : 0.0f);
    float xo = x_src[o];
    float xn = (xo + TAU_C * (dh + dv) + LT_C * img[o]) * RDEN_C;
    x_dst[o]  = xn;
    xt_dst[o] = 1.5f * xn - 0.5f * xo;         // x + theta*(x - x_old), theta=0.5
}

// =====================  Fallback: two kernels / iteration (3-plane ws)  =====================

// Dual step: pointwise in y => in-place safe. x_tilde tile (33x33) async-staged.
__global__ __launch_bounds__(1024)
void pd_dual(const float* __restrict__ w,
             const float* __restrict__ xt_src,
             const float* __restrict__ y_src,
             float* __restrict__ y_dst)
{
    constexpr int SX = TS + 1;                 // 33: own + right/down halo
    __shared__ float XT[SX * SX];
    const int tx  = threadIdx.x, ty = threadIdx.y;
    const int tid = ty * TS + tx;
    const int gx0 = blockIdx.x * TS;
    const int gy0 = blockIdx.y * TS;
    const size_t MN = (size_t)MM * NN;

    for (int e = tid; e < SX * SX; e += TS * TS) {
        int sy = e / SX, sx = e - sy * SX;
        int gy = clampi(gy0 + sy, 0, MM - 1);
        int gx = clampi(gx0 + sx, 0, NN - 1);
        async_g2l_b32(xt_src + (size_t)gy * NN + gx, &XT[e]);
    }
    wait_async_zero();
    __syncthreads();

    const int i = gy0 + ty, j = gx0 + tx;
    const size_t o = (size_t)i * NN + j;
    float w0 = w[o], w1 = w[MN + o];
    float xc = XT[ty * SX + tx];
    float xr = XT[ty * SX + tx + 1];
    float xd = XT[(ty + 1) * SX + tx];
    float y0n = fminf(fmaxf(y_src[o]      + SIGMA_C * (xr - xc) * w0, -1.0f), 1.0f);
    float y1n = fminf(fmaxf(y_src[MN + o] + SIGMA_C * (xd - xc) * w1, -1.0f), 1.0f);
    y_dst[o]      = y0n;
    y_dst[MN + o] = y1n;
}

// Primal step: stage h = w0*y0, v = w1*y1 (33x33 with top/left halo) in LDS.
__global__ __launch_bounds__(1024)
void pd_primal(const float* __restrict__ img,
               const float* __restrict__ w,
               const float* __restrict__ y,
               const float* __restrict__ x_src,
               float* __restrict__ x_dst,
               float* __restrict__ xt_dst)
{
    constexpr int EX = TS + 1;                 // 33
    __shared__ float H[EX * EX];
    __shared__ float V[EX * EX];
    const int tx  = threadIdx.x, ty = threadIdx.y;
    const int tid = ty * TS + tx;
    const int gx0 = blockIdx.x * TS;
    const int gy0 = blockIdx.y * TS;
    const size_t MN = (size_t)MM * NN;

    for (int e = tid; e < EX * EX; e += TS * TS) {
        int ey = e / EX, ex = e - ey * EX;
        int rg = gy0 - 1 + ey; rg = rg < 0 ? 0 : rg;
        int cg = gx0 - 1 + ex; cg = cg < 0 ? 0 : cg;
        size_t off = (size_t)rg * NN + cg;
        H[e] = w[off]      * y[off];
        V[e] = w[MN + off] * y[MN + off];
    }
    __syncthreads();

    const int i = gy0 + ty, j = gx0 + tx;
    const size_t o = (size_t)i * NN + j;
    float hC = H[(ty + 1) * EX + tx + 1];
    float hL = H[(ty + 1) * EX + tx];
    float vC = V[(ty + 1) * EX + tx + 1];
    float vU = V[ty * EX + tx + 1];
    float dh = (j < NN - 1 ? hC : 0.0f) - (j > 0 ? hL : 0.0f);
    float dv = (i < MM - 1 ? vC : 0.0f) - (i > 0 ? vU : 0.0f);
    float xo = x_src[o];
    float xn = (xo + TAU_C * (dh + dv) + LT_C * img[o]) * RDEN_C;
    x_dst[o]  = xn;
    xt_dst[o] = 1.5f * xn - 0.5f * xo;
}

extern "C" void kernel_launch(void* const* d_in, const int* in_sizes, int n_in,
                              void* d_out, int out_size, void* d_ws, size_t ws_size,
                              hipStream_t stream)
{
    const float* img = (const float*)d_in[0];   // [1, M, N]
    const float* w   = (const float*)d_in[1];   // [2, M, N]
    const float* y0  = (const float*)d_in[2];   // [2, M, N]
    float* out = (float*)d_out;                 // final x_tilde [1, M, N]

    const size_t MN = (size_t)MM * NN;
    char* ws = (char*)d_ws;
    dim3 grid(NN / TS, MM / TS);
    dim3 block(TS, TS);

    if (ws_size >= 6 * MN * sizeof(float)) {
        // Fused path: 1 kernel / iteration. Needs ping-pong for x_tilde and y.
        float* xb  = (float*)(ws);                // x           (1 plane)
        float* xtB = (float*)(ws + 1 * MN * 4);   // x_tilde odd (1 plane)
        float* yA  = (float*)(ws + 2 * MN * 4);   // y ping      (2 planes)
        float* yB  = (float*)(ws + 4 * MN * 4);   // y pong      (2 planes)
        // Even-iteration x_tilde lives in d_out, so iter 20 lands the result there.
        const float* xts = img;                   // x_tilde(0) == img_obs
        const float* ys  = y0;
        const float* xs  = img;                   // x(0) == img_obs
        for (int k = 1; k <= MAXIT; ++k) {
            float* yd  = (k & 1) ? yA : yB;
            float* xtd = (k & 1) ? xtB : out;
            pd_fused<<<grid, block, 0, stream>>>(img, w, xts, ys, xs, yd, xb, xtd);
            xts = xtd;
            ys  = yd;
            xs  = xb;                             // x updates in place after iter 1
        }
    } else {
        // Fallback: 2 kernels / iteration, 3 scratch planes (x + y), x_tilde in d_out.
        float* xb = (float*)(ws);                 // x (1 plane)
        float* yb = (float*)(ws + 1 * MN * 4);    // y (2 planes)
        const float* xts = img;
        const float* ys  = y0;
        const float* xs  = img;
        for (int k = 1; k <= MAXIT; ++k) {
            pd_dual<<<grid, block, 0, stream>>>(w, xts, ys, yb);
            pd_primal<<<grid, block, 0, stream>>>(img, w, yb, xs, xb, out);
            xts = out;                            // dual(k+1) reads x_tilde(k)
            ys  = yb;                             // in-place thereafter (pointwise)
            xs  = xb;
        }
    }
}